// ReinforcePointExtractor_14267881358077
// MI455X (gfx1250) — compile-verified
//
#include <hip/hip_runtime.h>
#include <hip/hip_bf16.h>

typedef float v2f __attribute__((ext_vector_type(2)));
typedef float v8f __attribute__((ext_vector_type(8)));

#define NB    16        // batch
#define NC    1024      // NBFEAT
#define HF    70
#define WF    70
#define HWF   (HF*WF)
#define HC    64
#define WC    64
#define HWC   4096      // 64*64
#define ENC   64
#define NP    1024      // top-k P

// ---- output layout (floats, reference return order, flattened) ----
#define OFF_POINTS   ((size_t)0)                         // 16*1024*67
#define OFF_BATCH    (OFF_POINTS + (size_t)NB*NP*67)     // 16384
#define OFF_POS      (OFF_BATCH  + (size_t)NB*NP)        // 16384*3
#define OFF_PFEAT    (OFF_POS    + (size_t)NB*NP*3)      // 16384*64
#define OFF_PROBS    (OFF_PFEAT  + (size_t)NB*NP*ENC)    // 16*4096
#define OFF_FLATIND  (OFF_PROBS  + (size_t)NB*HWC)       // 16*1024
#define OFF_BASEFEAT (OFF_FLATIND+ (size_t)NB*NP)        // 16*1024
#define OFF_BASELINE (OFF_BASEFEAT+(size_t)NB*NC)        // 16

// =====================================================================
// Kernel 1: fused 1x1-conv GEMM (WMMA f32 16x16x4) + conv bias
//           + baseFeat partial sums + lin projection + sigmoid + row sum
// grid = (32 p-groups, 16 batches), block = 256 (8 waves, wave32).
// Each wave: one 16-wide p-tile, all 64 output rows (4 WMMA o-tiles).
// =====================================================================
__global__ __launch_bounds__(256) void conv_wmma_fused(
    const float* __restrict__ fm,       // [16,1024,70,70]
    const float* __restrict__ conv_w,   // [64,1024]
    const float* __restrict__ conv_b,   // [64]
    const float* __restrict__ lin_w,    // [64]
    const float* __restrict__ lin_b,    // [1]
    float* __restrict__ pfm,            // ws: [16,64,4096]
    float* __restrict__ flatX,          // ws: [16,4096]
    float* __restrict__ sums,           // ws: [16]   (pre-zeroed)
    float* __restrict__ baseSum)        // ws: [16,1024] (pre-zeroed)
{
    __shared__ float ldsBase[NC];

    const int b    = blockIdx.y;
    const int tid  = threadIdx.x;
    const int wave = tid >> 5;
    const int lane = tid & 31;
    const int n    = lane & 15;       // N column within tile
    const int hi   = lane >> 4;       // 0: lanes 0-15, 1: lanes 16-31

    for (int i = tid; i < NC; i += 256) ldsBase[i] = 0.0f;
    __syncthreads();

    const int p = blockIdx.x * 128 + wave * 16 + n;   // 0..4095
    const int h = p >> 6, w = p & 63;
    // cropped fm element for channel c lives at fmb[c*HWF]
    const float* fmb = fm + (size_t)b * NC * HWF + (size_t)(h + 3) * WF + (w + 3);

    v8f acc0 = {}, acc1 = {}, acc2 = {}, acc3 = {};

    for (int c = 0; c < NC; c += 4) {
        const int kl = c + 2 * hi;                    // this half-wave's K pair
        // ---- B fragment (4x16 of fm), rows kl, kl+1 ----
        float b0 = fmb[(size_t)kl       * HWF];
        float b1 = fmb[(size_t)(kl + 1) * HWF];
        v2f bfrag; bfrag[0] = b0; bfrag[1] = b1;

        // ---- fused baseFeat: reduce the 16 columns of each loaded row ----
        float s0 = b0, s1 = b1;
        s0 += __shfl_xor(s0, 1, 32);  s1 += __shfl_xor(s1, 1, 32);
        s0 += __shfl_xor(s0, 2, 32);  s1 += __shfl_xor(s1, 2, 32);
        s0 += __shfl_xor(s0, 4, 32);  s1 += __shfl_xor(s1, 4, 32);
        s0 += __shfl_xor(s0, 8, 32);  s1 += __shfl_xor(s1, 8, 32);
        if (n == 0) {
            atomicAdd(&ldsBase[kl],     s0);
            atomicAdd(&ldsBase[kl + 1], s1);
        }

        // ---- A fragments + 4 WMMAs (o-tiles 0..3) ----
        {
            v2f a = *(const v2f*)(conv_w + (size_t)( 0 + n) * NC + kl);
            acc0 = __builtin_amdgcn_wmma_f32_16x16x4_f32(false, a, false, bfrag,
                                                         (short)0, acc0, false, false);
        }
        {
            v2f a = *(const v2f*)(conv_w + (size_t)(16 + n) * NC + kl);
            acc1 = __builtin_amdgcn_wmma_f32_16x16x4_f32(false, a, false, bfrag,
                                                         (short)0, acc1, false, false);
        }
        {
            v2f a = *(const v2f*)(conv_w + (size_t)(32 + n) * NC + kl);
            acc2 = __builtin_amdgcn_wmma_f32_16x16x4_f32(false, a, false, bfrag,
                                                         (short)0, acc2, false, false);
        }
        {
            v2f a = *(const v2f*)(conv_w + (size_t)(48 + n) * NC + kl);
            acc3 = __builtin_amdgcn_wmma_f32_16x16x4_f32(false, a, false, bfrag,
                                                         (short)0, acc3, false, false);
        }
    }

    // ---- epilogue: conv bias, pfm store, fused lin projection ----
    float xp = 0.0f;
    #pragma unroll
    for (int r = 0; r < 8; ++r) {
        int m0 = r + 8 * hi;                 // row within o-tile (C/D layout)
        {
            int m = m0;        float v = acc0[r] + conv_b[m];
            pfm[(((size_t)b * ENC + m) << 12) + p] = v;  xp += lin_w[m] * v;
        }
        {
            int m = 16 + m0;   float v = acc1[r] + conv_b[m];
            pfm[(((size_t)b * ENC + m) << 12) + p] = v;  xp += lin_w[m] * v;
        }
        {
            int m = 32 + m0;   float v = acc2[r] + conv_b[m];
            pfm[(((size_t)b * ENC + m) << 12) + p] = v;  xp += lin_w[m] * v;
        }
        {
            int m = 48 + m0;   float v = acc3[r] + conv_b[m];
            pfm[(((size_t)b * ENC + m) << 12) + p] = v;  xp += lin_w[m] * v;
        }
    }
    // combine the two half-wave partial dots (lane L and L^16 hold same N)
    float xo = xp + __shfl_xor(xp, 16, 32);
    float x  = xo + lin_b[0];
    float fx = 1.0f / (1.0f + expf(-x));     // sigmoid (duplicated in both halves)
    if (hi == 0) flatX[(size_t)b * HWC + p] = fx;
    // per-wave sum of 16 sigmoid values (reduce within 16-lane group)
    float sr = fx;
    sr += __shfl_xor(sr, 1, 32);
    sr += __shfl_xor(sr, 2, 32);
    sr += __shfl_xor(sr, 4, 32);
    sr += __shfl_xor(sr, 8, 32);
    if (lane == 0) atomicAdd(&sums[b], sr);

    // ---- flush baseFeat partials ----
    __syncthreads();
    for (int i = tid; i < NC; i += 256)
        atomicAdd(&baseSum[(size_t)b * NC + i], ldsBase[i]);
}

// =====================================================================
// Kernel 2: probs = flatX / (sum + eps), written straight to d_out
// =====================================================================
__global__ __launch_bounds__(256) void probs_kernel(
    const float* __restrict__ flatX, const float* __restrict__ sums,
    float* __restrict__ out)
{
    int b = blockIdx.y;
    int p = blockIdx.x * 256 + threadIdx.x;
    out[OFF_PROBS + (size_t)b * HWC + p] =
        flatX[(size_t)b * HWC + p] / (sums[b] + 1e-6f);
}

// =====================================================================
// Kernel 3: per-batch top-1024 of 4096 via LDS bitonic sort
// (descending by prob, stable ties by lower index -> matches lax.top_k)
// =====================================================================
__global__ __launch_bounds__(1024) void topk_kernel(
    const float* __restrict__ out_probs_region,   // d_out base
    float* __restrict__ out, int* __restrict__ wsInds)
{
    __shared__ float v[HWC];
    __shared__ int   id[HWC];
    const int b = blockIdx.x;
    for (int i = threadIdx.x; i < HWC; i += 1024) {
        v[i]  = out_probs_region[OFF_PROBS + (size_t)b * HWC + i];
        id[i] = i;
    }
    __syncthreads();
    for (int k = 2; k <= HWC; k <<= 1) {
        for (int j = k >> 1; j > 0; j >>= 1) {
            for (int i = threadIdx.x; i < HWC; i += 1024) {
                int ixj = i ^ j;
                if (ixj > i) {
                    float a = v[i], c = v[ixj];
                    int  ia = id[i], ic = id[ixj];
                    bool up = ((i & k) == 0);
                    // rank order: bigger value first; equal -> smaller index first
                    bool cFirst = (c > a) || (c == a && ic < ia);
                    bool aFirst = (a > c) || (a == c && ia < ic);
                    if (up ? cFirst : aFirst) {
                        v[i] = c; v[ixj] = a; id[i] = ic; id[ixj] = ia;
                    }
                }
            }
            __syncthreads();
        }
    }
    int i = threadIdx.x;          // first 1024 entries = top-k
    int fi = id[i];
    out[OFF_FLATIND + (size_t)b * NP + i] = (float)fi;
    wsInds[(size_t)b * NP + i] = fi;
}

// =====================================================================
// Kernel 4: gather point features; write points / batch / pos / pointfeatures
// block = 256 -> 4 points x 64 feature lanes
// =====================================================================
__global__ __launch_bounds__(256) void gather_kernel(
    const float* __restrict__ pfm, const int* __restrict__ wsInds,
    float* __restrict__ out)
{
    int g = blockIdx.x * 4 + (threadIdx.x >> 6);  // global point id 0..16383
    int e = threadIdx.x & 63;                     // feature channel
    int b = g >> 10;
    int fi = wsInds[g];
    float feat = pfm[(((size_t)b * ENC + e) << 12) + fi];
    out[OFF_POINTS + (size_t)g * 67 + 3 + e] = feat;
    out[OFF_PFEAT  + (size_t)g * 64 + e]     = feat;
    if (e < 3) {
        int a = fi & 63, o = fi >> 6;
        float vv = (e == 0) ? (float)a : (e == 1) ? (float)o : 0.0f;
        out[OFF_POINTS + (size_t)g * 67 + e] = vv;
        out[OFF_POS    + (size_t)g * 3  + e] = vv;
    }
    if (e == 0) out[OFF_BATCH + g] = (float)b;
}

// =====================================================================
// Kernel 5: baseFeat finalize (/4096) + baseline = relu(baseFeat . w + b)
// =====================================================================
__global__ __launch_bounds__(256) void baseline_kernel(
    const float* __restrict__ baseSum, const float* __restrict__ base_w,
    const float* __restrict__ base_b, float* __restrict__ out)
{
    __shared__ float red[256];
    int b = blockIdx.x, t = threadIdx.x;
    float dot = 0.0f;
    for (int c = t; c < NC; c += 256) {
        float bf = baseSum[(size_t)b * NC + c] * (1.0f / 4096.0f);
        out[OFF_BASEFEAT + (size_t)b * NC + c] = bf;
        dot += bf * base_w[c];
    }
    red[t] = dot; __syncthreads();
    for (int s = 128; s > 0; s >>= 1) {
        if (t < s) red[t] += red[t + s];
        __syncthreads();
    }
    if (t == 0) {
        float y = red[0] + base_b[0];
        out[OFF_BASELINE + b] = y > 0.0f ? y : 0.0f;
    }
}

extern "C" void kernel_launch(void* const* d_in, const int* in_sizes, int n_in,
                              void* d_out, int out_size, void* d_ws, size_t ws_size,
                              hipStream_t stream) {
    const float* fm     = (const float*)d_in[0];
    const float* conv_w = (const float*)d_in[1];
    const float* conv_b = (const float*)d_in[2];
    const float* lin_w  = (const float*)d_in[3];
    const float* lin_b  = (const float*)d_in[4];
    const float* base_w = (const float*)d_in[5];
    const float* base_b = (const float*)d_in[6];
    float* out = (float*)d_out;

    // workspace carve-up (floats)
    float* pfm     = (float*)d_ws;                        // 16*64*4096
    float* flatX   = pfm   + (size_t)NB * ENC * HWC;      // 16*4096
    float* sums    = flatX + (size_t)NB * HWC;            // 16
    float* baseSum = sums  + NB;                          // 16*1024
    int*   wsInds  = (int*)(baseSum + (size_t)NB * NC);   // 16*1024

    // zero the atomic accumulators (sums + baseSum are contiguous)
    hipMemsetAsync(sums, 0, (size_t)(NB + NB * NC) * sizeof(float), stream);

    conv_wmma_fused<<<dim3(32, NB), 256, 0, stream>>>(
        fm, conv_w, conv_b, lin_w, lin_b, pfm, flatX, sums, baseSum);
    probs_kernel<<<dim3(HWC / 256, NB), 256, 0, stream>>>(flatX, sums, out);
    topk_kernel<<<NB, 1024, 0, stream>>>(out, out, wsInds);
    gather_kernel<<<(NB * NP) / 4, 256, 0, stream>>>(pfm, wsInds, out);
    baseline_kernel<<<NB, 256, 0, stream>>>(baseSum, base_w, base_b, out);
}